// GridGraphImgNet_68650757259642
// MI455X (gfx1250) — compile-verified
//
#include <hip/hip_runtime.h>
#include <math.h>

// ---- problem dims (from reference) ----
#define N_NODES 32768
#define DHID    192
#define E_EDGES 262144
#define BGR     256
#define NPGR    128
#define DEG     8
#define EPG     (NPGR * DEG)     // 1024 edges per graph (contiguous)
#define EXPH    32
#define EXPD    128
#define QHD     128
#define QSD     256
#define OUTD    256
#define PDIM    1024
#define NCL     6
#define DINP    9
#define CSPLIT  2
#define CHALF   (DHID / CSPLIT)  // 96 channels per scatter block

typedef __attribute__((ext_vector_type(2))) float v2f;
typedef __attribute__((ext_vector_type(8))) float v8f;

// -------------------------------------------------------------------------
// Generic fp32 WMMA GEMM: D = op(A[MxK] * B[KxN] (+bias) (+extra)) one wave
// per 16x16 tile, K in steps of 4 via V_WMMA_F32_16X16X4_F32.
// flags: 1=add bias[col], 2=relu, 4=add extra[row*N+col]
// -------------------------------------------------------------------------
__global__ __launch_bounds__(32)
void gemm_wmma_f32(const float* __restrict__ A, const float* __restrict__ Bm,
                   const float* __restrict__ bias, const float* __restrict__ extra,
                   float* __restrict__ D, int M, int Nc, int K, int flags)
{
    const int m0   = blockIdx.x * 16;
    const int n0   = blockIdx.y * 16;
    const int lane = threadIdx.x;
    const int hi   = lane >> 4;      // 0 or 1
    const int lo   = lane & 15;

    v8f acc = {0.f, 0.f, 0.f, 0.f, 0.f, 0.f, 0.f, 0.f};
    const float* arow = A + (size_t)(m0 + lo) * K;

    for (int k0 = 0; k0 < K; k0 += 4) {
        const int ka = k0 + 2 * hi;
        v2f a, b;
        a.x = arow[ka];
        a.y = arow[ka + 1];
        b.x = Bm[(size_t)ka * Nc + n0 + lo];
        b.y = Bm[(size_t)(ka + 1) * Nc + n0 + lo];
        acc = __builtin_amdgcn_wmma_f32_16x16x4_f32(
            /*neg_a=*/false, a, /*neg_b=*/false, b,
            /*c_mod=*/(short)0, acc, /*reuse_a=*/false, /*reuse_b=*/false);
    }

#pragma unroll
    for (int r = 0; r < 8; ++r) {
        const int row = m0 + r + 8 * hi;
        const int col = n0 + lo;
        float v = acc[r];
        if (flags & 1) v += bias[col];
        if (flags & 4) v += extra[(size_t)row * Nc + col];
        if (flags & 2) v = fmaxf(v, 0.f);
        D[(size_t)row * Nc + col] = v;
    }
}

// -------------------------------------------------------------------------
// h0 = node_inputs[N,9] @ W_enc[9,192] + b_enc   (tiny K, plain VALU)
// -------------------------------------------------------------------------
__global__ void enc_kernel(const float* __restrict__ ni, const float* __restrict__ W,
                           const float* __restrict__ bb, float* __restrict__ h0)
{
    int idx = blockIdx.x * blockDim.x + threadIdx.x;
    if (idx >= N_NODES * DHID) return;
    int i = idx / DHID, j = idx % DHID;
    float acc = bb[j];
#pragma unroll
    for (int k = 0; k < DINP; ++k) acc += ni[i * DINP + k] * W[k * DHID + j];
    h0[idx] = acc;
}

// -------------------------------------------------------------------------
// Per-graph edge aggregation in LDS (ds_add_f32), no global atomics.
// Graph b owns nodes [b*128,(b+1)*128) and edges [b*1024,(b+1)*1024).
// Each block handles CHALF=96 channels -> 128*96*4 = 48KB LDS.
// agg[dst, ch..ch+95] = sum over edges of m[src, ch..ch+95]
// -------------------------------------------------------------------------
__global__ __launch_bounds__(256)
void scatter_lds_kernel(const float* __restrict__ m, const int* __restrict__ src,
                        const int* __restrict__ dst, float* __restrict__ agg)
{
    __shared__ float sagg[NPGR * CHALF];   // 49152 bytes
    const int b        = blockIdx.x;
    const int ch       = blockIdx.y * CHALF;
    const int nodebase = b * NPGR;
    const int ebase    = b * EPG;

    for (int i = threadIdx.x; i < NPGR * CHALF; i += 256) sagg[i] = 0.f;
    __syncthreads();

    for (int t = threadIdx.x; t < EPG * CHALF; t += 256) {
        const int e = ebase + t / CHALF;
        const int c = t % CHALF;
        const int s = src[e];
        const int d = dst[e] - nodebase;
        atomicAdd(&sagg[d * CHALF + c], m[(size_t)s * DHID + ch + c]);
    }
    __syncthreads();

    for (int i = threadIdx.x; i < NPGR * CHALF; i += 256) {
        const int n = i / CHALF, c = i % CHALF;
        agg[(size_t)(nodebase + n) * DHID + ch + c] = sagg[i];
    }
}

// gated: e2[i] *= sigmoid(g2[i])
__global__ void gate_kernel(const float* __restrict__ g2, float* __restrict__ e2, size_t n)
{
    size_t i = (size_t)blockIdx.x * blockDim.x + threadIdx.x;
    if (i < n) e2[i] *= 1.f / (1.f + __expf(-g2[i]));
}

// graph_encoding: contiguous 128-node segments (graph_ids = repeat(arange(B),128))
__global__ void ge_kernel(const float* __restrict__ gated, float* __restrict__ ge)
{
    int b = blockIdx.x, c = threadIdx.x;   // 128 threads
    const float* gb = gated + (size_t)b * NPGR * EXPD;
    float acc = 0.f;
    for (int n = 0; n < NPGR; ++n) acc += gb[(size_t)n * EXPD + c];
    ge[b * EXPD + c] = acc;
}

// q1 = relu(concat(ge, pos) @ Wq1 + bq1)
__global__ void q1_kernel(const float* __restrict__ ge, const float* __restrict__ ra,
                          const float* __restrict__ W, const float* __restrict__ bb,
                          float* __restrict__ q1)
{
    int idx = blockIdx.x * blockDim.x + threadIdx.x;
    if (idx >= BGR * QHD) return;
    int b = idx / QHD, j = idx % QHD;
    float acc = bb[j];
    for (int k = 0; k < EXPD; ++k) acc += ge[b * EXPD + k] * W[k * QHD + j];
#pragma unroll
    for (int k = 0; k < 3; ++k) acc += ra[b * DINP + k] * W[(EXPD + k) * QHD + j];
    q1[idx] = fmaxf(acc, 0.f);
}

// q = q1 @ Wq2 + bq2
__global__ void q2_kernel(const float* __restrict__ q1, const float* __restrict__ W,
                          const float* __restrict__ bb, float* __restrict__ q)
{
    int idx = blockIdx.x * blockDim.x + threadIdx.x;
    if (idx >= BGR * QSD) return;
    int b = idx / QSD, j = idx % QSD;
    float acc = bb[j];
    for (int k = 0; k < QHD; ++k) acc += q1[b * QHD + k] * W[k * QSD + j];
    q[idx] = acc;
}

// fused attention: scores -> softmax -> weighted sum, one block per graph
__global__ __launch_bounds__(1024)
void attn_kernel(const float* __restrict__ x, const float* __restrict__ q,
                 float* __restrict__ img)
{
    __shared__ float sh_q[QSD];
    __shared__ float sh_s[PDIM];
    __shared__ float sh_p[PDIM];
    const int b = blockIdx.x;
    const int t = threadIdx.x;                    // 0..1023
    const float* xb = x + (size_t)b * PDIM * OUTD;

    if (t < QSD) sh_q[t] = q[b * QSD + t];
    __syncthreads();

    // phase 1: scores, wave-cooperative coalesced dot products
    const int wave = t >> 5, lane = t & 31;
    for (int p = wave; p < PDIM; p += 32) {
        const float* xp = xb + (size_t)p * OUTD;
        float acc = 0.f;
#pragma unroll
        for (int c = lane; c < OUTD; c += 32) acc += xp[c] * sh_q[c];
        for (int m = 16; m > 0; m >>= 1) acc += __shfl_xor(acc, m, 32);
        if (lane == 0) sh_s[p] = acc;
    }
    __syncthreads();

    // softmax over 1024 patches
    sh_p[t] = sh_s[t];
    __syncthreads();
    for (int s = 512; s > 0; s >>= 1) {
        if (t < s) sh_p[t] = fmaxf(sh_p[t], sh_p[t + s]);
        __syncthreads();
    }
    const float mx = sh_p[0];
    __syncthreads();
    const float ev = __expf(sh_s[t] - mx);
    sh_p[t] = ev;
    sh_s[t] = ev;
    __syncthreads();
    for (int s = 512; s > 0; s >>= 1) {
        if (t < s) sh_p[t] += sh_p[t + s];
        __syncthreads();
    }
    const float inv = 1.f / sh_p[0];
    __syncthreads();
    sh_s[t] *= inv;        // attn weights
    __syncthreads();

    // phase 2: img[c] = sum_p attn[p] * x[b,p,c]  (L2-resident re-read)
    const int c = t & 255, g = t >> 8;
    float acc = 0.f;
    for (int p = g; p < PDIM; p += 4) acc += sh_s[p] * xb[(size_t)p * OUTD + c];
    sh_p[t] = acc;
    __syncthreads();
    if (t < OUTD)
        img[b * OUTD + t] = sh_p[t] + sh_p[t + 256] + sh_p[t + 512] + sh_p[t + 768];
}

// c1 = relu(concat(img, ge) @ Wc1 + bc1)
__global__ void c1_kernel(const float* __restrict__ img, const float* __restrict__ ge,
                          const float* __restrict__ W, const float* __restrict__ bb,
                          float* __restrict__ c1)
{
    int idx = blockIdx.x * blockDim.x + threadIdx.x;
    if (idx >= BGR * 32) return;
    int b = idx / 32, j = idx % 32;
    float acc = bb[j];
    for (int k = 0; k < OUTD; ++k) acc += img[b * OUTD + k] * W[k * 32 + j];
    for (int k = 0; k < EXPD; ++k) acc += ge[b * EXPD + k] * W[(OUTD + k) * 32 + j];
    c1[idx] = fmaxf(acc, 0.f);
}

// classes = c1 @ Wc2 + bc2 ; write `nodes` output = [pos | classes]
__global__ void c2_kernel(const float* __restrict__ c1, const float* __restrict__ W,
                          const float* __restrict__ bb, const float* __restrict__ ra,
                          float* __restrict__ cls, float* __restrict__ out_nodes)
{
    int idx = blockIdx.x * blockDim.x + threadIdx.x;
    if (idx >= BGR * 9) return;
    int b = idx / 9, j = idx % 9;
    if (j < 3) {
        out_nodes[b * 9 + j] = ra[b * DINP + j];
    } else {
        int jj = j - 3;
        float acc = bb[jj];
        for (int k = 0; k < 32; ++k) acc += c1[b * 32 + k] * W[k * NCL + jj];
        cls[b * NCL + jj] = acc;
        out_nodes[b * 9 + j] = acc;
    }
}

// new_node_features = [pos | softmax(classes)] @ W_enc + b_enc
__global__ void nnf_kernel(const float* __restrict__ ra, const float* __restrict__ cls,
                           const float* __restrict__ W, const float* __restrict__ bb,
                           float* __restrict__ nnf)
{
    int idx = blockIdx.x * blockDim.x + threadIdx.x;
    if (idx >= BGR * DHID) return;
    int b = idx / DHID, j = idx % DHID;
    float mx = cls[b * NCL];
#pragma unroll
    for (int k = 1; k < NCL; ++k) mx = fmaxf(mx, cls[b * NCL + k]);
    float ex[NCL]; float s = 0.f;
#pragma unroll
    for (int k = 0; k < NCL; ++k) { ex[k] = __expf(cls[b * NCL + k] - mx); s += ex[k]; }
    float inv = 1.f / s;
    float acc = bb[j];
#pragma unroll
    for (int k = 0; k < 3; ++k) acc += ra[b * DINP + k] * W[k * DHID + j];
#pragma unroll
    for (int k = 0; k < NCL; ++k) acc += ex[k] * inv * W[(3 + k) * DHID + j];
    nnf[idx] = acc;
}

// edge MLP: one wave per node, lane = hidden unit (EH == wave32 == 32)
__global__ __launch_bounds__(128)
void edge_mlp_kernel(const float* __restrict__ h2, const float* __restrict__ nnf,
                     const float* __restrict__ img, const int* __restrict__ gids,
                     const float* __restrict__ W1, const float* __restrict__ b1,
                     const float* __restrict__ W2, const float* __restrict__ b2,
                     float* __restrict__ out)
{
    int node = (blockIdx.x * blockDim.x + threadIdx.x) >> 5;
    int lane = threadIdx.x & 31;
    if (node >= N_NODES) return;
    int gid = gids[node];
    const float* hrow = h2 + (size_t)node * DHID;
    const float* nrow = nnf + (size_t)gid * DHID;
    const float* irow = img + (size_t)gid * OUTD;
    float acc = b1[lane];
    for (int k = 0; k < DHID; ++k) acc += hrow[k] * W1[k * 32 + lane];
    for (int k = 0; k < DHID; ++k) acc += nrow[k] * W1[(DHID + k) * 32 + lane];
    for (int k = 0; k < OUTD; ++k) acc += irow[k] * W1[(2 * DHID + k) * 32 + lane];
    float hv = fmaxf(acc, 0.f) * W2[lane];
    for (int m = 16; m > 0; m >>= 1) hv += __shfl_xor(hv, m, 32);
    if (lane == 0) out[node] = 1.f / (1.f + __expf(-(hv + b2[0])));
}

// -------------------------------------------------------------------------
extern "C" void kernel_launch(void* const* d_in, const int* in_sizes, int n_in,
                              void* d_out, int out_size, void* d_ws, size_t ws_size,
                              hipStream_t stream)
{
    const float* node_inputs = (const float*)d_in[0];
    const float* ref_atoms   = (const float*)d_in[1];
    const float* x_encs      = (const float*)d_in[2];
    const int*   edges       = (const int*)d_in[3];
    const int*   graph_ids   = (const int*)d_in[4];
    const float* W_enc = (const float*)d_in[5];  const float* b_enc = (const float*)d_in[6];
    const float* Wn    = (const float*)d_in[7];  const float* bn    = (const float*)d_in[8];
    const float* Wm    = (const float*)d_in[9];
    const float* We1   = (const float*)d_in[10]; const float* be1   = (const float*)d_in[11];
    const float* We2   = (const float*)d_in[12]; const float* be2   = (const float*)d_in[13];
    const float* Wg1   = (const float*)d_in[14]; const float* bg1   = (const float*)d_in[15];
    const float* Wg2   = (const float*)d_in[16]; const float* bg2   = (const float*)d_in[17];
    const float* Wq1   = (const float*)d_in[18]; const float* bq1   = (const float*)d_in[19];
    const float* Wq2   = (const float*)d_in[20]; const float* bq2   = (const float*)d_in[21];
    const float* Wc1   = (const float*)d_in[22]; const float* bc1   = (const float*)d_in[23];
    const float* Wc2   = (const float*)d_in[24]; const float* bc2   = (const float*)d_in[25];
    const float* Wed1  = (const float*)d_in[26]; const float* bed1  = (const float*)d_in[27];
    const float* Wed2  = (const float*)d_in[28]; const float* bed2  = (const float*)d_in[29];

    const int* e_src = edges;
    const int* e_dst = edges + E_EDGES;

    float* out_nodes = (float*)d_out;            // [256*9]
    float* out_edges = (float*)d_out + BGR * 9;  // [32768]

    // ---- workspace layout (floats), phase-ordered aliasing ----
    const size_t NB = (size_t)N_NODES * DHID;    // 6,291,456
    float* ws     = (float*)d_ws;
    float* bufA   = ws;            // h0, later h2
    float* bufM   = ws + NB;       // m / e2(gated) / m2
    float* bufAGG = ws + 2 * NB;   // agg / {e1,g1,g2} / agg2
    float* bufH   = ws + 3 * NB;   // h
    float* e1 = bufAGG;
    float* g1 = bufAGG + (size_t)N_NODES * EXPH;
    float* g2 = bufAGG + 2 * (size_t)N_NODES * EXPH;
    float* e2 = bufM;
    float* sm   = ws + 4 * NB;
    float* ge   = sm;                    // 256*128
    float* q1   = ge + BGR * EXPD;       // 256*128
    float* qv   = q1 + BGR * QHD;        // 256*256
    float* img  = qv + BGR * QSD;        // 256*256
    float* c1   = img + BGR * OUTD;      // 256*32
    float* cls  = c1 + BGR * 32;         // 256*6
    float* nnf  = cls + BGR * NCL;       // 256*192

    // 1) h0 = node_inputs @ W_enc + b_enc
    enc_kernel<<<(N_NODES * DHID + 255) / 256, 256, 0, stream>>>(node_inputs, W_enc, b_enc, bufA);

    // ---- GNN layer 1 ----
    gemm_wmma_f32<<<dim3(N_NODES / 16, DHID / 16), 32, 0, stream>>>(bufA, Wm, nullptr, nullptr, bufM, N_NODES, DHID, DHID, 0);
    scatter_lds_kernel<<<dim3(BGR, CSPLIT), 256, 0, stream>>>(bufM, e_src, e_dst, bufAGG);
    gemm_wmma_f32<<<dim3(N_NODES / 16, DHID / 16), 32, 0, stream>>>(bufA, Wn, bn, bufAGG, bufH, N_NODES, DHID, DHID, 1 | 2 | 4);

    // ---- gated experts ----
    gemm_wmma_f32<<<dim3(N_NODES / 16, EXPH / 16), 32, 0, stream>>>(bufH, We1, be1, nullptr, e1, N_NODES, EXPH, DHID, 1 | 2);
    gemm_wmma_f32<<<dim3(N_NODES / 16, EXPD / 16), 32, 0, stream>>>(e1, We2, be2, nullptr, e2, N_NODES, EXPD, EXPH, 1);
    gemm_wmma_f32<<<dim3(N_NODES / 16, EXPH / 16), 32, 0, stream>>>(bufH, Wg1, bg1, nullptr, g1, N_NODES, EXPH, DHID, 1 | 2);
    gemm_wmma_f32<<<dim3(N_NODES / 16, EXPD / 16), 32, 0, stream>>>(g1, Wg2, bg2, nullptr, g2, N_NODES, EXPD, EXPH, 1);
    gate_kernel<<<(unsigned)(((size_t)N_NODES * EXPD + 255) / 256), 256, 0, stream>>>(g2, e2, (size_t)N_NODES * EXPD);
    ge_kernel<<<BGR, EXPD, 0, stream>>>(e2, ge);

    // ---- query + attention ----
    q1_kernel<<<(BGR * QHD + 255) / 256, 256, 0, stream>>>(ge, ref_atoms, Wq1, bq1, q1);
    q2_kernel<<<(BGR * QSD + 255) / 256, 256, 0, stream>>>(q1, Wq2, bq2, qv);
    attn_kernel<<<BGR, 1024, 0, stream>>>(x_encs, qv, img);

    // ---- classifier + nodes output + new node features ----
    c1_kernel<<<(BGR * 32 + 255) / 256, 256, 0, stream>>>(img, ge, Wc1, bc1, c1);
    c2_kernel<<<(BGR * 9 + 255) / 256, 256, 0, stream>>>(c1, Wc2, bc2, ref_atoms, cls, out_nodes);
    nnf_kernel<<<(BGR * DHID + 255) / 256, 256, 0, stream>>>(ref_atoms, cls, W_enc, b_enc, nnf);

    // ---- GNN layer 2 ----
    gemm_wmma_f32<<<dim3(N_NODES / 16, DHID / 16), 32, 0, stream>>>(bufH, Wm, nullptr, nullptr, bufM, N_NODES, DHID, DHID, 0);
    scatter_lds_kernel<<<dim3(BGR, CSPLIT), 256, 0, stream>>>(bufM, e_src, e_dst, bufAGG);
    gemm_wmma_f32<<<dim3(N_NODES / 16, DHID / 16), 32, 0, stream>>>(bufH, Wn, bn, bufAGG, bufA, N_NODES, DHID, DHID, 1 | 2 | 4);

    // ---- edge MLP -> new_edges output ----
    edge_mlp_kernel<<<N_NODES / 4, 128, 0, stream>>>(bufA, nnf, img, graph_ids,
                                                     Wed1, bed1, Wed2, bed2, out_edges);
}